// DeepSetLayer_87110526697916
// MI455X (gfx1250) — compile-verified
//
#include <hip/hip_runtime.h>
#include <math.h>

typedef float v2f __attribute__((ext_vector_type(2)));
typedef float v8f __attribute__((ext_vector_type(8)));

#define N_NODES 50000
#define N_EDGES 600000
#define E_TOT   (N_EDGES + N_NODES)
#define IN_F    128
#define OUT_F   128
#define SMALL   12
#define INV_DK  0.28867513459f   // 1/sqrt(12)
#define MT      4                // node tiles (of 16) per block in fused kernel
#define MROWS   (16 * MT)

// ---------------------------------------------------------------------------
// K0: repack W1, W2 into K-paired, N-major layout:
//     Wp[((k>>1)*OUT_F + n)*2 + (k&1)] = W[n*IN_F + k]
//     so a WMMA B-fragment (K = ka, ka+1 at lane n) is ONE 8-byte load,
//     coalesced across lanes.
// ---------------------------------------------------------------------------
__global__ __launch_bounds__(256) void repack_w_kernel(
    const float* __restrict__ W1, const float* __restrict__ W2,
    float* __restrict__ W1p, float* __restrict__ W2p)
{
    int i = blockIdx.x * 256 + threadIdx.x;   // over OUT_F*IN_F
    if (i >= OUT_F * IN_F) return;
    int n = i / IN_F;       // out feature (row of W)
    int k = i % IN_F;       // in  feature (col of W)
    int o = ((k >> 1) * OUT_F + n) * 2 + (k & 1);
    W1p[o] = W1[i];
    W2p[o] = W2[i];
}

// ---------------------------------------------------------------------------
// K1: q = tanh(node @ Wq^T + bq), k = node @ Wk^T + bk   (one thread / node)
// ---------------------------------------------------------------------------
__global__ __launch_bounds__(256) void qk_kernel(
    const float* __restrict__ node_data,
    const float* __restrict__ Wq, const float* __restrict__ bq,
    const float* __restrict__ Wk, const float* __restrict__ bk,
    float* __restrict__ qout, float* __restrict__ kout)
{
    __shared__ float sWq[SMALL][IN_F];
    __shared__ float sWk[SMALL][IN_F];
    __shared__ float sbq[SMALL];
    __shared__ float sbk[SMALL];

    const int tid = threadIdx.x;
    for (int i = tid; i < SMALL * IN_F; i += 256) {
        sWq[i / IN_F][i % IN_F] = Wq[i];
        sWk[i / IN_F][i % IN_F] = Wk[i];
    }
    if (tid < SMALL) { sbq[tid] = bq[tid]; sbk[tid] = bk[tid]; }
    __syncthreads();

    const int n = blockIdx.x * 256 + tid;
    if (n >= N_NODES) return;

    float aq[SMALL], ak[SMALL];
#pragma unroll
    for (int j = 0; j < SMALL; ++j) { aq[j] = 0.f; ak[j] = 0.f; }

    const float4* nd = (const float4*)(node_data + (size_t)n * IN_F);
    for (int c = 0; c < IN_F / 4; ++c) {
        float4 x = nd[c];
        int k0 = c * 4;
#pragma unroll
        for (int j = 0; j < SMALL; ++j) {
            aq[j] += x.x * sWq[j][k0]     + x.y * sWq[j][k0 + 1]
                   + x.z * sWq[j][k0 + 2] + x.w * sWq[j][k0 + 3];
            ak[j] += x.x * sWk[j][k0]     + x.y * sWk[j][k0 + 1]
                   + x.z * sWk[j][k0 + 2] + x.w * sWk[j][k0 + 3];
        }
    }
#pragma unroll
    for (int j = 0; j < SMALL; ++j) {
        qout[(size_t)n * SMALL + j] = tanhf(aq[j] + sbq[j]);
        kout[(size_t)n * SMALL + j] = ak[j] + sbk[j];
    }
}

// ---------------------------------------------------------------------------
// K2: per-edge score -> w = exp(score/sqrt(d)); hw-atomic denom[dst] += w
//     edges [0,E) are real, [E, E+N) are self loops (s=d=e-E)
// ---------------------------------------------------------------------------
__global__ __launch_bounds__(256) void score_kernel(
    const float* __restrict__ q, const float* __restrict__ k,
    const int* __restrict__ src, const int* __restrict__ dst,
    float* __restrict__ wbuf, float* __restrict__ denom)
{
    int e = blockIdx.x * 256 + threadIdx.x;
    if (e >= E_TOT) return;
    int s, d;
    if (e < N_EDGES) { s = src[e]; d = dst[e]; }
    else             { s = e - N_EDGES; d = s; }

    const float4* qp = (const float4*)(q + (size_t)s * SMALL); // 48B stride -> 16B aligned
    const float4* kp = (const float4*)(k + (size_t)d * SMALL);
    float sc = 0.f;
#pragma unroll
    for (int i = 0; i < 3; ++i) {
        float4 a = qp[i], b = kp[i];
        sc += a.x * b.x + a.y * b.y + a.z * b.z + a.w * b.w;
    }
    float w = expf(sc * INV_DK);
    wbuf[e] = w;
    unsafeAtomicAdd(denom + d, w);   // global_atomic_add_f32
}

// ---------------------------------------------------------------------------
// K3: att[d] += w[e] * node_data[s]   (one wave per edge, 4 features/lane)
// ---------------------------------------------------------------------------
__global__ __launch_bounds__(256) void aggregate_kernel(
    const float* __restrict__ node_data,
    const int* __restrict__ src, const int* __restrict__ dst,
    const float* __restrict__ wbuf, float* __restrict__ att)
{
    int gid = blockIdx.x * 256 + threadIdx.x;
    int e = gid >> 5;
    if (e >= E_TOT) return;
    int lane = gid & 31;
    int s, d;
    if (e < N_EDGES) { s = src[e]; d = dst[e]; }
    else             { s = e - N_EDGES; d = s; }

    float w = wbuf[e];
    int f = lane * 4;
    float4 v = *(const float4*)(node_data + (size_t)s * IN_F + f);
    float* p = att + (size_t)d * IN_F + f;
    unsafeAtomicAdd(p + 0, w * v.x);
    unsafeAtomicAdd(p + 1, w * v.y);
    unsafeAtomicAdd(p + 2, w * v.z);
    unsafeAtomicAdd(p + 3, w * v.w);
}

// ---------------------------------------------------------------------------
// K4: x = node @ W1^T + (att/denom) @ W2^T + b2 ; row-L2-normalize ; ReLU
//     Block = 64 nodes x 128 out features; 8 waves, each covers a 16-outf
//     column strip across MT=4 row tiles, reusing each B fragment 4x.
//     fp32 WMMA: V_WMMA_F32_16X16X4_F32, K-loop step 4.
// ---------------------------------------------------------------------------
__global__ __launch_bounds__(256) void fused_out_kernel(
    const float* __restrict__ node_data,
    const float* __restrict__ att,       // unnormalized
    const float* __restrict__ denom,
    const float* __restrict__ W1p,       // K-paired layout (see repack)
    const float* __restrict__ W2p,
    const float* __restrict__ b2,
    float* __restrict__ out)
{
    __shared__ float sA[MROWS][IN_F];    // node tile
    __shared__ float sT[MROWS][IN_F];    // normalized att tile
    __shared__ float sPart[8][MROWS];    // per-wave partial row sums of squares
    __shared__ float sRn[MROWS];         // 1/||row||

    const int tid   = threadIdx.x;
    const int wave  = tid >> 5;
    const int lane  = tid & 31;
    const int half  = lane >> 4;         // 0: lanes 0-15, 1: lanes 16-31
    const int l16   = lane & 15;
    const int node0 = blockIdx.x * MROWS;
    const int nbase = wave * 16;         // out-feature base for this wave

    // stage 64x128 node + att tiles (att normalized by denom), float4 loads
    for (int i = tid; i < MROWS * (IN_F / 4); i += 256) {
        int m = i >> 5;                  // / (IN_F/4)
        int f = (i & 31) * 4;
        int node = node0 + m;
        float4 nd = make_float4(0.f, 0.f, 0.f, 0.f);
        float4 av = make_float4(0.f, 0.f, 0.f, 0.f);
        if (node < N_NODES) {
            nd = *(const float4*)(node_data + (size_t)node * IN_F + f);
            float4 a = *(const float4*)(att + (size_t)node * IN_F + f);
            float inv = 1.0f / denom[node];
            av = make_float4(a.x * inv, a.y * inv, a.z * inv, a.w * inv);
        }
        *(float4*)&sA[m][f] = nd;
        *(float4*)&sT[m][f] = av;
    }
    __syncthreads();

    v8f acc[MT];
#pragma unroll
    for (int t = 0; t < MT; ++t) acc[t] = (v8f){};

    // A frag (16x4 f32): lane holds M = lane&15; VGPR0/1 = K = 2*half + {0,1}
    // B frag (4x16 f32): lane holds N = lane&15; VGPR0/1 = K = 2*half + {0,1}
    for (int k0 = 0; k0 < IN_F; k0 += 4) {
        const int ka = k0 + 2 * half;                 // even
        const int bo = ((ka >> 1) * OUT_F + nbase + l16) * 2;
        v2f b1 = *(const v2f*)(W1p + bo);             // one b64, coalesced
        v2f b2v = *(const v2f*)(W2p + bo);
#pragma unroll
        for (int t = 0; t < MT; ++t) {
            const int m = t * 16 + l16;
            v2f a1 = *(const v2f*)&sA[m][ka];
            acc[t] = __builtin_amdgcn_wmma_f32_16x16x4_f32(
                         false, a1, false, b1, (short)0, acc[t], false, false);
            v2f a2 = *(const v2f*)&sT[m][ka];
            acc[t] = __builtin_amdgcn_wmma_f32_16x16x4_f32(
                         false, a2, false, b2v, (short)0, acc[t], false, false);
        }
    }

    // + bias (depends only on n)
    const float bias = b2[nbase + l16];
#pragma unroll
    for (int t = 0; t < MT; ++t)
#pragma unroll
        for (int j = 0; j < 8; ++j) acc[t][j] += bias;

    // row sum-of-squares: acc[t] VGPR j, half h holds (m = t*16 + j + 8h, n = l16)
#pragma unroll
    for (int t = 0; t < MT; ++t) {
#pragma unroll
        for (int j = 0; j < 8; ++j) {
            float sq = acc[t][j] * acc[t][j];
            sq += __shfl_xor(sq, 1, 32);
            sq += __shfl_xor(sq, 2, 32);
            sq += __shfl_xor(sq, 4, 32);
            sq += __shfl_xor(sq, 8, 32); // reduced within each 16-lane half
            if (l16 == 0) sPart[wave][t * 16 + j + 8 * half] = sq;
        }
    }
    __syncthreads();
    if (tid < MROWS) {
        float s = 0.f;
#pragma unroll
        for (int w = 0; w < 8; ++w) s += sPart[w][tid];
        sRn[tid] = 1.0f / sqrtf(s);
    }
    __syncthreads();

#pragma unroll
    for (int t = 0; t < MT; ++t) {
#pragma unroll
        for (int j = 0; j < 8; ++j) {
            int m = t * 16 + j + 8 * half;
            int node = node0 + m;
            float v = acc[t][j] * sRn[m];
            v = fmaxf(v, 0.f);
            if (node < N_NODES) out[(size_t)node * OUT_F + nbase + l16] = v;
        }
    }
}

// ---------------------------------------------------------------------------
extern "C" void kernel_launch(void* const* d_in, const int* in_sizes, int n_in,
                              void* d_out, int out_size, void* d_ws, size_t ws_size,
                              hipStream_t stream)
{
    const float* node_data = (const float*)d_in[0];
    const int*   src       = (const int*)  d_in[1];
    const int*   dst       = (const int*)  d_in[2];
    const float* Wq        = (const float*)d_in[3];
    const float* bq        = (const float*)d_in[4];
    const float* Wk        = (const float*)d_in[5];
    const float* bk        = (const float*)d_in[6];
    const float* W1        = (const float*)d_in[7];
    const float* W2        = (const float*)d_in[8];
    const float* b2        = (const float*)d_in[9];
    float*       out       = (float*)d_out;

    float* ws    = (float*)d_ws;
    float* q     = ws;                                   // N*12
    float* kbuf  = q     + (size_t)N_NODES * SMALL;      // N*12
    float* W1p   = kbuf  + (size_t)N_NODES * SMALL;      // 128*128
    float* W2p   = W1p   + IN_F * OUT_F;                 // 128*128
    float* wbuf  = W2p   + IN_F * OUT_F;                 // E_TOT
    float* denom = wbuf  + E_TOT;                        // N
    float* att   = denom + N_NODES;                      // N*128

    hipMemsetAsync(denom, 0, (size_t)N_NODES * sizeof(float), stream);
    hipMemsetAsync(att,   0, (size_t)N_NODES * IN_F * sizeof(float), stream);

    repack_w_kernel<<<(OUT_F * IN_F + 255) / 256, 256, 0, stream>>>(W1, W2, W1p, W2p);
    qk_kernel<<<(N_NODES + 255) / 256, 256, 0, stream>>>(node_data, Wq, bq, Wk, bk, q, kbuf);
    score_kernel<<<(E_TOT + 255) / 256, 256, 0, stream>>>(q, kbuf, src, dst, wbuf, denom);
    aggregate_kernel<<<(E_TOT * 32 + 255) / 256, 256, 0, stream>>>(node_data, src, dst, wbuf, att);
    fused_out_kernel<<<(N_NODES + MROWS - 1) / MROWS, 256, 0, stream>>>(
        node_data, att, denom, W1p, W2p, b2, out);
}